// STRNNCell_9895604650719
// MI455X (gfx1250) — compile-verified
//
#include <hip/hip_runtime.h>
#include <math.h>

#define Hdim 512
#define Ldim 4096

typedef __attribute__((ext_vector_type(2))) float v2f;
typedef __attribute__((ext_vector_type(8))) float v8f;

#define WMMA_F32(A, B, C) \
    __builtin_amdgcn_wmma_f32_16x16x4_f32(false, (A), false, (B), (short)0, (C), false, false)

// ---------------------------------------------------------------------------
// Kernel 1: v = rowcombine_td( emb @ W_th_upper^T , emb @ W_th_lower^T )
// emb rows gathered on the fly from loc_table[loc[i]].
// Block = 256 threads (8 waves). Block tile: 64 (M) x 128 (N).
// Each wave owns a 2x2 grid of 16x16 C tiles per weight matrix (8 accums),
// so every A and B LDS fragment feeds two WMMAs (6 loads -> 8 wmma).
// ---------------------------------------------------------------------------
__global__ __launch_bounds__(256) void k_gemm1(
    const float* __restrict__ td_upper, const float* __restrict__ td_lower,
    const int*   __restrict__ loc,
    const float* __restrict__ Wtu, const float* __restrict__ Wtl,
    const float* __restrict__ loc_table,
    float* __restrict__ v_out)
{
    __shared__ float sA [ 64][36];   // emb tile [m][k]  (stride 36: 16B-aligned rows, conflict-free)
    __shared__ float sBu[128][36];   // Wtu tile [n][k]  (B[k][n] = Wtu[n][k])
    __shared__ float sBl[128][36];   // Wtl tile [n][k]

    const int tid  = threadIdx.x;
    const int gm0  = blockIdx.x * 64;      // row block in L
    const int gn0  = blockIdx.y * 128;     // col block in H
    const int wave = tid >> 5;
    const int lane = tid & 31;
    const int half = lane >> 4;            // 0: lanes 0-15, 1: lanes 16-31
    const int lr   = lane & 15;

    const int tm0 = (wave & 1) * 2;        // C tile rows tm0, tm0+1   (0..3)
    const int tn0 = (wave >> 1) * 2;       // C tile cols tn0, tn0+1   (0..7)

    // float4 staging: thread covers rows (tid>>3) and (tid>>3)+32 of sA.
    const int sr = tid >> 3;               // 0..31
    const int sc = (tid & 7) * 4;          // 0,4,...,28
    long long rowoff[2];
    rowoff[0] = (long long)loc[gm0 + sr] * Hdim;
    rowoff[1] = (long long)loc[gm0 + sr + 32] * Hdim;

    v8f aU00 = {}, aU01 = {}, aU10 = {}, aU11 = {};
    v8f aL00 = {}, aL01 = {}, aL10 = {}, aL11 = {};

    for (int kb = 0; kb < Hdim; kb += 32) {
#pragma unroll
        for (int p = 0; p < 2; ++p) {
            *(float4*)&sA[sr + p * 32][sc] =
                *(const float4*)&loc_table[rowoff[p] + kb + sc];
        }
#pragma unroll
        for (int p = 0; p < 4; ++p) {
            int r = sr + p * 32;           // 0..127
            *(float4*)&sBu[r][sc] = *(const float4*)&Wtu[(gn0 + r) * Hdim + kb + sc];
            *(float4*)&sBl[r][sc] = *(const float4*)&Wtl[(gn0 + r) * Hdim + kb + sc];
        }
        __syncthreads();

#pragma unroll
        for (int k4 = 0; k4 < 32; k4 += 4) {
            int kk = k4 + 2 * half;        // even column -> 8B-aligned b64 fragment load
            v2f a0  = *(const v2f*)&sA [ tm0      * 16 + lr][kk];
            v2f a1  = *(const v2f*)&sA [(tm0 + 1) * 16 + lr][kk];
            v2f bu0 = *(const v2f*)&sBu[ tn0      * 16 + lr][kk];
            v2f bu1 = *(const v2f*)&sBu[(tn0 + 1) * 16 + lr][kk];
            v2f bl0 = *(const v2f*)&sBl[ tn0      * 16 + lr][kk];
            v2f bl1 = *(const v2f*)&sBl[(tn0 + 1) * 16 + lr][kk];
            aU00 = WMMA_F32(a0, bu0, aU00);
            aU01 = WMMA_F32(a0, bu1, aU01);
            aL00 = WMMA_F32(a0, bl0, aL00);
            aL01 = WMMA_F32(a0, bl1, aL01);
            aU10 = WMMA_F32(a1, bu0, aU10);
            aU11 = WMMA_F32(a1, bu1, aU11);
            aL10 = WMMA_F32(a1, bl0, aL10);
            aL11 = WMMA_F32(a1, bl1, aL11);
        }
        __syncthreads();
    }

    // Row-wise combine with td scalars and store v tiles.
#pragma unroll
    for (int r = 0; r < 8; ++r) {
        int row0 = gm0 + tm0 * 16 + r + 8 * half;   // C layout: VGPR r -> M = r + 8*half
        int row1 = row0 + 16;
        float tu0 = td_upper[row0], tl0 = td_lower[row0];
        float tu1 = td_upper[row1], tl1 = td_lower[row1];
        float i0 = 1.0f / (tu0 + tl0);
        float i1 = 1.0f / (tu1 + tl1);
        int col0 = gn0 + tn0 * 16 + lr;
        v_out[row0 * Hdim + col0]      = (aU00[r] * tu0 + aL00[r] * tl0) * i0;
        v_out[row0 * Hdim + col0 + 16] = (aU01[r] * tu0 + aL01[r] * tl0) * i0;
        v_out[row1 * Hdim + col0]      = (aU10[r] * tu1 + aL10[r] * tl1) * i1;
        v_out[row1 * Hdim + col0 + 16] = (aU11[r] * tu1 + aL11[r] * tl1) * i1;
    }
}

// ---------------------------------------------------------------------------
// Kernel 2: w = rowcombine_ld( v @ W_sh_upper^T , v @ W_sh_lower^T );
// deterministic in-block sum of the 64x128 w tile -> partials[block].
// ---------------------------------------------------------------------------
__global__ __launch_bounds__(256) void k_gemm2(
    const float* __restrict__ ld_upper, const float* __restrict__ ld_lower,
    const float* __restrict__ v_in,
    const float* __restrict__ Wsu, const float* __restrict__ Wsl,
    float* __restrict__ partials)
{
    __shared__ float sA [ 64][36];
    __shared__ float sBu[128][36];
    __shared__ float sBl[128][36];
    __shared__ float red[256];

    const int tid  = threadIdx.x;
    const int gm0  = blockIdx.x * 64;
    const int gn0  = blockIdx.y * 128;
    const int wave = tid >> 5;
    const int lane = tid & 31;
    const int half = lane >> 4;
    const int lr   = lane & 15;
    const int tm0  = (wave & 1) * 2;
    const int tn0  = (wave >> 1) * 2;
    const int sr   = tid >> 3;
    const int sc   = (tid & 7) * 4;

    v8f aU00 = {}, aU01 = {}, aU10 = {}, aU11 = {};
    v8f aL00 = {}, aL01 = {}, aL10 = {}, aL11 = {};

    for (int kb = 0; kb < Hdim; kb += 32) {
#pragma unroll
        for (int p = 0; p < 2; ++p) {
            int r = sr + p * 32;
            *(float4*)&sA[r][sc] = *(const float4*)&v_in[(gm0 + r) * Hdim + kb + sc];
        }
#pragma unroll
        for (int p = 0; p < 4; ++p) {
            int r = sr + p * 32;
            *(float4*)&sBu[r][sc] = *(const float4*)&Wsu[(gn0 + r) * Hdim + kb + sc];
            *(float4*)&sBl[r][sc] = *(const float4*)&Wsl[(gn0 + r) * Hdim + kb + sc];
        }
        __syncthreads();

#pragma unroll
        for (int k4 = 0; k4 < 32; k4 += 4) {
            int kk = k4 + 2 * half;
            v2f a0  = *(const v2f*)&sA [ tm0      * 16 + lr][kk];
            v2f a1  = *(const v2f*)&sA [(tm0 + 1) * 16 + lr][kk];
            v2f bu0 = *(const v2f*)&sBu[ tn0      * 16 + lr][kk];
            v2f bu1 = *(const v2f*)&sBu[(tn0 + 1) * 16 + lr][kk];
            v2f bl0 = *(const v2f*)&sBl[ tn0      * 16 + lr][kk];
            v2f bl1 = *(const v2f*)&sBl[(tn0 + 1) * 16 + lr][kk];
            aU00 = WMMA_F32(a0, bu0, aU00);
            aU01 = WMMA_F32(a0, bu1, aU01);
            aL00 = WMMA_F32(a0, bl0, aL00);
            aL01 = WMMA_F32(a0, bl1, aL01);
            aU10 = WMMA_F32(a1, bu0, aU10);
            aU11 = WMMA_F32(a1, bu1, aU11);
            aL10 = WMMA_F32(a1, bl0, aL10);
            aL11 = WMMA_F32(a1, bl1, aL11);
        }
        __syncthreads();
    }

    float p = 0.0f;
#pragma unroll
    for (int r = 0; r < 8; ++r) {
        int row0 = gm0 + tm0 * 16 + r + 8 * half;
        int row1 = row0 + 16;
        float lu0 = ld_upper[row0], ll0 = ld_lower[row0];
        float lu1 = ld_upper[row1], ll1 = ld_lower[row1];
        float i0 = 1.0f / (lu0 + ll0);
        float i1 = 1.0f / (lu1 + ll1);
        p += (aU00[r] * lu0 + aL00[r] * ll0) * i0;
        p += (aU01[r] * lu0 + aL01[r] * ll0) * i0;
        p += (aU10[r] * lu1 + aL10[r] * ll1) * i1;
        p += (aU11[r] * lu1 + aL11[r] * ll1) * i1;
    }

    red[tid] = p;
    __syncthreads();
    for (int s = 128; s > 0; s >>= 1) {
        if (tid < s) red[tid] += red[tid + s];
        __syncthreads();
    }
    if (tid == 0) partials[blockIdx.y * gridDim.x + blockIdx.x] = red[0];
}

// ---------------------------------------------------------------------------
// Kernel 3: loc_vec = sum(partials[256]); usr = W_ih @ hx;
// out = sigmoid(loc_vec + usr)
// ---------------------------------------------------------------------------
__global__ __launch_bounds__(512) void k_final(
    const float* __restrict__ partials,
    const float* __restrict__ W_ih, const float* __restrict__ hx,
    float* __restrict__ out)
{
    __shared__ float red[512];
    __shared__ float shx[512];
    const int t = threadIdx.x;
    red[t] = (t < 256) ? partials[t] : 0.0f;
    shx[t] = hx[t];
    __syncthreads();
    for (int s = 256; s > 0; s >>= 1) {
        if (t < s) red[t] += red[t + s];
        __syncthreads();
    }
    float loc_sum = red[0];

    float acc = 0.0f;
    const float* wrow = W_ih + t * Hdim;
    for (int k = 0; k < Hdim; k += 4) {
        acc += wrow[k]     * shx[k];
        acc += wrow[k + 1] * shx[k + 1];
        acc += wrow[k + 2] * shx[k + 2];
        acc += wrow[k + 3] * shx[k + 3];
    }
    out[t] = 1.0f / (1.0f + __expf(-(loc_sum + acc)));
}

// ---------------------------------------------------------------------------
extern "C" void kernel_launch(void* const* d_in, const int* in_sizes, int n_in,
                              void* d_out, int out_size, void* d_ws, size_t ws_size,
                              hipStream_t stream) {
    const float* td_u      = (const float*)d_in[0];
    const float* td_l      = (const float*)d_in[1];
    const float* ld_u      = (const float*)d_in[2];
    const float* ld_l      = (const float*)d_in[3];
    const int*   loc       = (const int*)  d_in[4];
    const float* hx        = (const float*)d_in[5];
    const float* W_ih      = (const float*)d_in[6];
    const float* W_tu      = (const float*)d_in[7];
    const float* W_tl      = (const float*)d_in[8];
    const float* W_su      = (const float*)d_in[9];
    const float* W_sl      = (const float*)d_in[10];
    const float* loc_table = (const float*)d_in[11];
    float* out = (float*)d_out;

    float* partials = (float*)d_ws;                       // 256 floats
    float* v_buf    = (float*)((char*)d_ws + 4096);       // L*H floats (8 MB)

    dim3 grid(Ldim / 64, Hdim / 128);                     // (64, 4) = 256 blocks
    k_gemm1<<<grid, 256, 0, stream>>>(td_u, td_l, loc, W_tu, W_tl, loc_table, v_buf);
    k_gemm2<<<grid, 256, 0, stream>>>(ld_u, ld_l, v_buf, W_su, W_sl, partials);
    k_final<<<1, 512, 0, stream>>>(partials, W_ih, hx, out);
}